// CascadeRCNN_73701638800032
// MI455X (gfx1250) — compile-verified
//
#include <hip/hip_runtime.h>
#include <hip/hip_bf16.h>
#include <stdint.h>

// ---------------- problem constants (from reference) ----------------
#define IMG_WF      1280.0f
#define IMG_HF      1280.0f
#define NANCH       230400        // 160*160*9
#define NBATCH      4
#define MAX_DETS    1000
#define IOU_TH      0.5f
#define DW_CLAMPF   4.0f

// ---------------- NMS partitioning ----------------
#define NWG_PER_IMG 16
#define STRIPE      (NANCH / NWG_PER_IMG)   // 14400 anchors per WGP -> 288KB LDS
#define NMS_THREADS 512                      // 16 wave32 per WG

// workspace layout (bytes)
#define SCORES_BYTES ((size_t)NBATCH * NANCH * 4)          //  3,686,400
#define BOXES_BYTES  ((size_t)NBATCH * NANCH * 16)         // 14,745,600
#define CAND_BYTES   ((size_t)2 * NBATCH * NWG_PER_IMG * 8)

typedef unsigned int u32x4 __attribute__((ext_vector_type(4)));
typedef unsigned int u32x8 __attribute__((ext_vector_type(8)));

// =====================================================================
// Phase 1: decode + sigmoid + clip + validity  (bandwidth bound, ~2us)
// =====================================================================
__global__ __launch_bounds__(256)
void decode_kernel(const float* __restrict__ obj,
                   const float* __restrict__ deltas,
                   const float* __restrict__ anchors,
                   float* __restrict__ ws_scores,
                   float4* __restrict__ ws_boxes) {
    int i = blockIdx.x * blockDim.x + threadIdx.x;
    if (i >= NBATCH * NANCH) return;
    float4 d = ((const float4*)deltas)[i];
    float4 a = ((const float4*)anchors)[i];
    float  o = obj[i];

    float w  = a.z - a.x, h = a.w - a.y;
    float cx = a.x + 0.5f * w, cy = a.y + 0.5f * h;
    float px = d.x * w + cx,   py = d.y * h + cy;
    float pw = __expf(fminf(d.z, DW_CLAMPF)) * w;
    float ph = __expf(fminf(d.w, DW_CLAMPF)) * h;

    float x1 = fminf(fmaxf(px - 0.5f * pw, 0.0f), IMG_WF);
    float y1 = fminf(fmaxf(py - 0.5f * ph, 0.0f), IMG_HF);
    float x2 = fminf(fmaxf(px + 0.5f * pw, 0.0f), IMG_WF);
    float y2 = fminf(fmaxf(py + 0.5f * ph, 0.0f), IMG_HF);

    bool valid = (x2 - x1 >= 1.0f) && (y2 - y1 >= 1.0f);
    // sigmoid with raw v_rcp_f32 (no IEEE-div fixup chain needed here)
    float s = __builtin_amdgcn_rcpf(1.0f + __expf(-o));

    ws_scores[i] = valid ? s : -1.0f;
    ws_boxes[i]  = make_float4(x1, y1, x2, y2);
}

// =====================================================================
// Tiny init: zero the inter-WG barrier state (deterministic every call)
// =====================================================================
__global__ void init_sync_kernel(unsigned* sync) {
    if (threadIdx.x < 2 * NBATCH) sync[threadIdx.x] = 0u;
}

// =====================================================================
// Device-scope barrier among the 16 WGs of one image (sense via gen ctr)
// =====================================================================
__device__ __forceinline__ void img_barrier(unsigned* cnt, unsigned* gen) {
    __syncthreads();
    if (threadIdx.x == 0) {
        __threadfence();
        unsigned g = __hip_atomic_load(gen, __ATOMIC_ACQUIRE, __HIP_MEMORY_SCOPE_AGENT);
        if (atomicAdd(cnt, 1u) == (unsigned)(NWG_PER_IMG - 1)) {
            __hip_atomic_store(cnt, 0u, __ATOMIC_RELAXED, __HIP_MEMORY_SCOPE_AGENT);
            __threadfence();
            atomicAdd(gen, 1u);
        } else {
            while (__hip_atomic_load(gen, __ATOMIC_ACQUIRE, __HIP_MEMORY_SCOPE_AGENT) == g) {
                __builtin_amdgcn_s_sleep(1);
            }
        }
        __threadfence();
    }
    __syncthreads();
}

// =====================================================================
// Phase 2: greedy NMS, one image striped across 16 WGPs.
// Entire stripe (scores + boxes) resident in 288KB of CDNA5 LDS.
// Stage-in uses both CDNA5 async engines: TDM (tensor_load_to_lds,
// TENSORcnt) for the contiguous box stripe, and per-lane
// global_load_async_to_lds_* (ASYNCcnt) as the data path.
// =====================================================================
__global__ __launch_bounds__(NMS_THREADS)
void nms_kernel(const float4* __restrict__ boxes_g,
                const float* __restrict__ scores_g,
                unsigned long long* __restrict__ cand,   // [2][B][NWG]
                unsigned* __restrict__ sync,             // [B] cnt, [B] gen
                float* __restrict__ out_boxes,           // [B][MAX_DETS][4]
                float* __restrict__ out_scores,          // [B][MAX_DETS]
                float* __restrict__ out_valid)           // [B][MAX_DETS]
{
    __shared__ float4 sBox[STRIPE];                 // 230,400 B
    __shared__ float  sSc[STRIPE];                  //  57,600 B
    __shared__ float  sRedS[NMS_THREADS / 32];
    __shared__ int    sRedI[NMS_THREADS / 32];
    __shared__ float  sBcast[6];

    const int tid  = threadIdx.x;
    const int b    = blockIdx.x / NWG_PER_IMG;
    const int wg   = blockIdx.x % NWG_PER_IMG;
    const int base = wg * STRIPE;

    const float4* gb = boxes_g  + (size_t)b * NANCH + base;
    const float*  gs = scores_g + (size_t)b * NANCH + base;

    // ---- TDM: one descriptor moves the whole 230,400B box stripe ----
    if (tid < 32) {   // wave 0 only issues the tensor DMA
        unsigned lds_off = (unsigned)(uintptr_t)(&sBox[0]);
        unsigned long long ga = (unsigned long long)(uintptr_t)gb;
        u32x4 g0;
        g0[0] = 1u;                                      // count=1 (valid user D#)
        g0[1] = lds_off;                                 // lds_addr [63:32]
        g0[2] = (unsigned)(ga & 0xffffffffu);            // global_addr[31:0]
        g0[3] = (unsigned)((ga >> 32) & 0x01ffffffu)     // global_addr[56:32]
                | 0x80000000u;                           // type=2 ("image")
        u32x8 g1;
        g1[0] = 0x00020000u;   // data_size=2 (4B); no mask/pad/iterate
        g1[1] = 0xE1000000u;   // tensor_dim0 = 57600 (0xE100) low16 in [31:16]
        g1[2] = 0x00010000u;   // tensor_dim0 hi16=0 ; tensor_dim1 = 1
        g1[3] = 0xE1000000u;   // tensor_dim1 hi16=0 ; tile_dim0 = 57600
        g1[4] = 0x00000001u;   // tile_dim1 = 1 ; tile_dim2 = 0
        g1[5] = 0x0000E100u;   // tensor_dim0_stride = 57600 (low 32)
        g1[6] = 0u;            // stride hi ; tensor_dim1_stride lo
        g1[7] = 0u;
        asm volatile("tensor_load_to_lds %0, %1" :: "s"(g0), "s"(g1) : "memory");
        asm volatile("s_wait_tensorcnt 0x0" ::: "memory");
    }

    // ---- per-lane async copy engine: scores + boxes -> LDS ----
    for (int i = tid; i < STRIPE; i += NMS_THREADS) {
        unsigned lb = (unsigned)(uintptr_t)(&sBox[i]);   // low 32 bits = LDS offset
        unsigned ls = (unsigned)(uintptr_t)(&sSc[i]);
        const void* ga  = (const void*)(gb + i);
        const void* gsa = (const void*)(gs + i);
        asm volatile("global_load_async_to_lds_b128 %0, %1, off"
                     :: "v"(lb), "v"(ga) : "memory");
        asm volatile("global_load_async_to_lds_b32 %0, %1, off"
                     :: "v"(ls), "v"(gsa) : "memory");
    }
    asm volatile("s_wait_asynccnt 0" ::: "memory");
    __syncthreads();

    unsigned* cnt = sync + b;
    unsigned* gen = sync + NBATCH + b;

    for (int t = 0; t < MAX_DETS; ++t) {
        // ---- local argmax over this WG's LDS stripe ----
        float bs = -2.0f; int bi = 0;
        for (int i = tid; i < STRIPE; i += NMS_THREADS) {
            float s = sSc[i];
            if (s > bs) { bs = s; bi = i; }
        }
        // wave32 shuffle reduce
        for (int off = 16; off > 0; off >>= 1) {
            float os = __shfl_down(bs, off, 32);
            int   oi = __shfl_down(bi, off, 32);
            if (os > bs) { bs = os; bi = oi; }
        }
        if ((tid & 31) == 0) { sRedS[tid >> 5] = bs; sRedI[tid >> 5] = bi; }
        __syncthreads();
        if (tid == 0) {
            float ms = -2.0f; int mi = 0;
            for (int w = 0; w < NMS_THREADS / 32; ++w)
                if (sRedS[w] > ms) { ms = sRedS[w]; mi = sRedI[w]; }
            unsigned long long pk =
                (((unsigned long long)__float_as_uint(ms)) << 32) |
                (unsigned)(base + mi);
            cand[((size_t)(t & 1) * NBATCH + b) * NWG_PER_IMG + wg] = pk;
        }

        // ---- exchange candidates across the 16 WGPs of this image ----
        img_barrier(cnt, gen);

        if (tid == 0) {
            float ws = -2.0f; unsigned wi = 0;
            const unsigned long long* cb =
                cand + ((size_t)(t & 1) * NBATCH + b) * NWG_PER_IMG;
            for (int k = 0; k < NWG_PER_IMG; ++k) {
                unsigned long long p = cb[k];
                float s = __uint_as_float((unsigned)(p >> 32));
                if (s > ws) { ws = s; wi = (unsigned)(p & 0xffffffffu); }
            }
            float4 wb = boxes_g[(size_t)b * NANCH + wi];
            sBcast[0] = __uint_as_float(wi);
            sBcast[1] = ws;
            sBcast[2] = wb.x; sBcast[3] = wb.y; sBcast[4] = wb.z; sBcast[5] = wb.w;
            if (wg == 0) {                       // one writer per image
                float m = (ws > 0.0f) ? 1.0f : 0.0f;
                float* ob = out_boxes + ((size_t)b * MAX_DETS + t) * 4;
                ob[0] = wb.x * m; ob[1] = wb.y * m;
                ob[2] = wb.z * m; ob[3] = wb.w * m;
                out_scores[(size_t)b * MAX_DETS + t] = (ws > 0.0f) ? ws : 0.0f;
                out_valid [(size_t)b * MAX_DETS + t] = m;
            }
        }
        __syncthreads();

        unsigned wi = __float_as_uint(sBcast[0]);
        float ws  = sBcast[1];
        float wx1 = sBcast[2], wy1 = sBcast[3], wx2 = sBcast[4], wy2 = sBcast[5];
        bool selok = ws > 0.0f;

        if (!selok) {
            // everything exhausted: remaining outputs are zeros; all 16 WGs
            // of this image see the same ws and break together.
            if (wg == 0) {
                for (int tt = t + 1 + tid; tt < MAX_DETS; tt += NMS_THREADS) {
                    float* ob = out_boxes + ((size_t)b * MAX_DETS + tt) * 4;
                    ob[0] = 0.0f; ob[1] = 0.0f; ob[2] = 0.0f; ob[3] = 0.0f;
                    out_scores[(size_t)b * MAX_DETS + tt] = 0.0f;
                    out_valid [(size_t)b * MAX_DETS + tt] = 0.0f;
                }
            }
            break;
        }

        float warea = fmaxf(wx2 - wx1, 0.0f) * fmaxf(wy2 - wy1, 0.0f);
        int wlocal = (int)wi - base;             // may be outside [0,STRIPE)

        // ---- suppression pass over the LDS stripe ----
        for (int i = tid; i < STRIPE; i += NMS_THREADS) {
            if (i == wlocal) { sSc[i] = -1.0f; continue; }  // ms[idx] = -1
            float s = sSc[i];
            if (s <= -1.0f) continue;                        // already dead
            float4 bx = sBox[i];
            float ix1 = fmaxf(wx1, bx.x), iy1 = fmaxf(wy1, bx.y);
            float ix2 = fminf(wx2, bx.z), iy2 = fminf(wy2, bx.w);
            float inter = fmaxf(ix2 - ix1, 0.0f) * fmaxf(iy2 - iy1, 0.0f);
            float area  = fmaxf(bx.z - bx.x, 0.0f) * fmaxf(bx.w - bx.y, 0.0f);
            float iou   = inter / (warea + area - inter + 1e-9f);
            if (iou > IOU_TH) sSc[i] = -1.0f;
        }
        __syncthreads();
    }
}

// =====================================================================
extern "C" void kernel_launch(void* const* d_in, const int* in_sizes, int n_in,
                              void* d_out, int out_size, void* d_ws, size_t ws_size,
                              hipStream_t stream) {
    const float* obj     = (const float*)d_in[0];   // [B][N]
    const float* deltas  = (const float*)d_in[1];   // [B][N][4]
    const float* anchors = (const float*)d_in[2];   // [B][N][4]
    float* out = (float*)d_out;                     // boxes | scores | valid

    char* ws = (char*)d_ws;
    float*              ws_scores = (float*)ws;
    float4*             ws_boxes  = (float4*)(ws + SCORES_BYTES);
    unsigned long long* cand      = (unsigned long long*)(ws + SCORES_BYTES + BOXES_BYTES);
    unsigned*           syncw     = (unsigned*)(ws + SCORES_BYTES + BOXES_BYTES + CAND_BYTES);

    init_sync_kernel<<<1, 32, 0, stream>>>(syncw);

    int total = NBATCH * NANCH;
    decode_kernel<<<(total + 255) / 256, 256, 0, stream>>>(
        obj, deltas, anchors, ws_scores, ws_boxes);

    float* out_boxes  = out;
    float* out_scores = out + (size_t)NBATCH * MAX_DETS * 4;
    float* out_valid  = out + (size_t)NBATCH * MAX_DETS * 5;
    nms_kernel<<<NBATCH * NWG_PER_IMG, NMS_THREADS, 0, stream>>>(
        ws_boxes, ws_scores, cand, syncw, out_boxes, out_scores, out_valid);
}